// STGCN_71485435674706
// MI455X (gfx1250) — compile-verified
//
#include <hip/hip_runtime.h>

#define N_NODES 97
#define IN_T 12
#define HID 6
#define GATES 24

typedef float v2f __attribute__((ext_vector_type(2)));
typedef float v8f __attribute__((ext_vector_type(8)));

// D = A(16x4) * B(4x16) + C, fp32 WMMA (CDNA5)
__device__ __forceinline__ v8f wmma_f32(v2f a, v2f b, v8f c) {
  return __builtin_amdgcn_wmma_f32_16x16x4_f32(false, a, false, b, (short)0, c,
                                               false, false);
}

// ---- optional CDNA5 async global->LDS staging (guarded: falls back cleanly)
#if __has_builtin(__builtin_amdgcn_global_load_async_to_lds_b32)
#define USE_ASYNC_LDS 1
typedef __attribute__((address_space(1))) int as1_int;  // __device__ int*
typedef __attribute__((address_space(3))) int as3_int;  // __shared__ int*
__device__ __forceinline__ void async_ld_b32(const float* g, float* l) {
  // generic->LDS offset: low 32 bits of the generic address ARE the LDS offset
  __builtin_amdgcn_global_load_async_to_lds_b32(
      (as1_int*)(uintptr_t)g, (as3_int*)(uint32_t)(uintptr_t)l, 0, 0);
}
__device__ __forceinline__ void wait_async0() {
#if __has_builtin(__builtin_amdgcn_s_wait_asynccnt)
  __builtin_amdgcn_s_wait_asynccnt(0);
#else
  asm volatile("s_wait_asynccnt 0x0" ::: "memory");
#endif
}
#endif

// ---------------------------------------------------------------------------
// Kernel 1: x_proj = X(M x 12) * W_ih^T(12 x 24) + b_ih   (M = T*97)
// One wave: 16(M) x 32(N: 24 valid) tile, 3 k-steps. All loads unconditional
// (clamped addresses); garbage columns 24..31 are discarded at the store.
// ---------------------------------------------------------------------------
__global__ __launch_bounds__(128) void xproj_wmma(
    const float* __restrict__ X, const float* __restrict__ Wih,
    const float* __restrict__ bih, float* __restrict__ xp, long M) {
  const int l = threadIdx.x & 31;
  const int lr = l & 15;
  const int lh = l >> 4;
  const long wave = (long)blockIdx.x * 4 + (threadIdx.x >> 5);
  const long m0 = wave * 16;
  if (m0 >= M) return;  // wave-uniform exit
  long mrow = m0 + lr;
  if (mrow >= M) mrow = M - 1;  // clamp for partial tail tile
  const float* xrow = X + mrow * IN_T;
  const int n1 = 16 + lr;
  const int n1c = (n1 < GATES) ? n1 : (GATES - 1);
  v8f acc0 = {0.f, 0.f, 0.f, 0.f, 0.f, 0.f, 0.f, 0.f};
  v8f acc1 = {0.f, 0.f, 0.f, 0.f, 0.f, 0.f, 0.f, 0.f};
#pragma unroll
  for (int kk = 0; kk < IN_T; kk += 4) {
    const int ka = kk + lh * 2;  // lanes 0-15: K=kk,kk+1 ; 16-31: K=kk+2,kk+3
    v2f a, b0, b1;
    a.x = xrow[ka];
    a.y = xrow[ka + 1];
    b0.x = Wih[lr * IN_T + ka];
    b0.y = Wih[lr * IN_T + ka + 1];
    b1.x = Wih[n1c * IN_T + ka];
    b1.y = Wih[n1c * IN_T + ka + 1];
    acc0 = wmma_f32(a, b0, acc0);
    acc1 = wmma_f32(a, b1, acc1);
  }
  const float bv0 = bih[lr];   // hoisted: one load, reused by all 8 rows
  const float bv1 = bih[n1c];  // clamped -> unconditional
  if (m0 + 16 <= M) {          // wave-uniform fast path: full tile, no guards
    const long mb = m0 + lh * 8;  // C/D: lane<16 -> M=r ; lane>=16 -> M=8+r
#pragma unroll
    for (int r = 0; r < 8; ++r) xp[(mb + r) * GATES + lr] = acc0[r] + bv0;
    if (n1 < GATES) {  // lane-constant guard hoisted out of the row loop
#pragma unroll
      for (int r = 0; r < 8; ++r) xp[(mb + r) * GATES + n1] = acc1[r] + bv1;
    }
  } else {
#pragma unroll
    for (int r = 0; r < 8; ++r) {
      const long m = m0 + r + lh * 8;
      if (m < M) {
        xp[m * GATES + lr] = acc0[r] + bv0;
        if (n1 < GATES) xp[m * GATES + n1] = acc1[r] + bv1;
      }
    }
  }
}

// ---------------------------------------------------------------------------
// Kernel 2: LSTM scan. One wave per node (97 independent serial chains).
// Lanes 0..23 = one gate row each. ALL lanes redundantly update chain lane%6
// (lanes 0..5 are canonical for the shuffles) so the serial critical path has
// no divergent control flow; only the store is lane-guarded.
// ---------------------------------------------------------------------------
__global__ __launch_bounds__(32) void lstm_scan(
    const float* __restrict__ xp, const float* __restrict__ Whh,
    const float* __restrict__ bhh_g, float* __restrict__ al, int T) {
  const int n = blockIdx.x;
  const int lane = threadIdx.x;
  float wh[HID];
  float bh = 0.f;
  if (lane < GATES) {
#pragma unroll
    for (int j = 0; j < HID; ++j) wh[j] = Whh[lane * HID + j];
    bh = bhh_g[lane];
  } else {
#pragma unroll
    for (int j = 0; j < HID; ++j) wh[j] = 0.f;
  }
  // i(0-5),f(6-11) sigmoid; g(12-17) tanh; o(18-23) sigmoid.
  // tanh(x) = 2*sigmoid(2x)-1 -> one exp per lane either way.
  const bool isTanh = (lane >= 12 && lane < 18);
  const float am = isTanh ? 2.f : 1.f;
  const float ac = isTanh ? -1.f : 0.f;
  const int j6 = lane % HID;
  const long stride = (long)N_NODES * GATES;
  const float* src = xp + (long)n * GATES + lane;

  float hs = 0.f, c = 0.f;
  float gcur = (lane < GATES) ? src[0] : 0.f;
  for (int t = 0; t < T; ++t) {
    float gnext = ((lane < GATES) && (t + 1 < T)) ? src[(long)(t + 1) * stride] : 0.f;
    if (t + 8 < T) __builtin_prefetch(src + (long)(t + 8) * stride, 0, 1);
    float g = gcur + bh;
#pragma unroll
    for (int j = 0; j < HID; ++j) g = fmaf(wh[j], __shfl(hs, j, 32), g);
    float sv = 1.f / (1.f + __expf(-am * g));
    float act = fmaf(am, sv, ac);
    const float iv = __shfl(act, j6, 32);
    const float fv = __shfl(act, j6 + HID, 32);
    const float gv = __shfl(act, j6 + 2 * HID, 32);
    const float ov = __shfl(act, j6 + 3 * HID, 32);
    // branch-free state update (every lane tracks chain j6)
    c = fmaf(fv, c, iv * gv);
    const float tc = 2.f / (1.f + __expf(-2.f * c)) - 1.f;
    const float hn = ov * tc;
    hs = 2.f / (1.f + __expf(-2.f * hn)) - 1.f;  // extra tanh, fed back
    if (lane < HID) al[((long)t * N_NODES + n) * HID + lane] = hs;
    gcur = gnext;
  }
}

// ---------------------------------------------------------------------------
// Kernel 3 (x3): 3-tap conv along H(=6) as WMMA GEMM.
//   in (T,CIN,6) -> out (T,COUT,6) = relu(conv + bias)
//   M rows = (t,h), K = CIN*3, N = COUT (only kw=1 of 3x3 contributes).
// Weight index is LINEAR in k: W[co][ci][kh][1] = W[co*9*CIN + 3k + 1].
// LDS tile [tl][ci][8] keeps a zero H-halo so im2col is a plain LDS read.
// Wave computes 16x32. The (q,r) im2col walk repeats with period 3 k-steps
// (LDS base +32 floats, weight ptr +36 floats): steady-state loop is manually
// unrolled x3 with lane-half-constant offsets (no selects, no div).
// STATS: fused per-channel sum/sumsq (BatchNorm) via LDS + global atomics.
// ---------------------------------------------------------------------------
template <int CIN, int COUT, bool STATS>
__global__ __launch_bounds__(128) void conv3x1_wmma(
    const float* __restrict__ in, const float* __restrict__ W,
    const float* __restrict__ bias, float* __restrict__ out,
    float* __restrict__ gsum, float* __restrict__ gsq, int T) {
  constexpr int TB = 8;
  constexpr int K = CIN * 3;
  constexpr int NFULL = K / 4;       // guard-free k-steps
  constexpr int N3 = NFULL / 3;      // steady-state unrolled-x3 iterations
  constexpr int MT = (TB * 6) / 16;  // 3
  constexpr int NT = (COUT + 15) / 16;
  constexpr int NTP = (NT + 1) / 2;    // 16x32 wave tiles
  __shared__ float lin[TB * CIN * 8];  // data at [1..6], zeros at 0 and 7
  __shared__ float bsum[NT * 16];
  __shared__ float bsq[NT * 16];

  const int t0 = blockIdx.x * TB;
  const int tid = threadIdx.x;

  // ---- stage input tile (division-free; t clamped so loads are valid) ----
  for (int tl = 0; tl < TB; ++tl) {
    const int tgc = min(t0 + tl, T - 1);
    for (int ci = tid; ci < CIN; ci += 128) {
      float* lrow = lin + (tl * CIN + ci) * 8;
      const float* grow = in + ((long)tgc * CIN + ci) * 6;
      lrow[0] = 0.f;
      lrow[7] = 0.f;
#if defined(USE_ASYNC_LDS)
#pragma unroll
      for (int e = 0; e < 6; ++e) async_ld_b32(grow + e, lrow + 1 + e);
#else
#pragma unroll
      for (int e = 0; e < 6; ++e) lrow[1 + e] = grow[e];
#endif
    }
  }
#if defined(USE_ASYNC_LDS)
  wait_async0();
#endif
  if (STATS) {
    for (int i = tid; i < NT * 16; i += 128) {
      bsum[i] = 0.f;
      bsq[i] = 0.f;
    }
  }
  __syncthreads();

  const int l = tid & 31, wv = tid >> 5;
  const int lr = l & 15, lh = l >> 4;
  const bool fullT = (t0 + TB <= T);  // wave-uniform

  // Per-lane-half constant LDS offsets for 3 consecutive k-steps.
  // k = 3q + r -> ofs = q*8 + r.  lanes 0-15 walk k=4i{+1}; 16-31 k=4i+2{+3}.
  const int oa0 = lh ? 2 : 0, ob0 = lh ? 8 : 1;
  const int oa1 = lh ? 16 : 9, ob1 = lh ? 17 : 10;
  const int oa2 = lh ? 25 : 18, ob2 = lh ? 26 : 24;

  for (int tile = wv; tile < MT * NTP; tile += 4) {  // wave-uniform loop
    const int mt = tile / NTP, ntp = tile % NTP;
    const int mloc = mt * 16 + lr;
    const int tl = mloc / 6, hh = mloc % 6;
    const int nn0 = ntp * 32 + lr;
    const int nn1 = nn0 + 16;
    const int nc0 = min(nn0, COUT - 1);  // clamp: garbage cols are discarded
    const int nc1 = min(nn1, COUT - 1);
    const float* lA = lin + tl * (CIN * 8) + hh;
    const float* b0p = W + (long)nc0 * CIN * 9 + 3 * (lh * 2) + 1;
    const float* b1p = W + (long)nc1 * CIN * 9 + 3 * (lh * 2) + 1;
    v8f acc0 = {0.f, 0.f, 0.f, 0.f, 0.f, 0.f, 0.f, 0.f};
    v8f acc1 = {0.f, 0.f, 0.f, 0.f, 0.f, 0.f, 0.f, 0.f};
    const float* la = lA;
    for (int i3 = 0; i3 < N3; ++i3) {  // 12 k-values per iteration
      v2f a, b0v, b1v;
      a.x = la[oa0];
      a.y = la[ob0];
      b0v.x = b0p[0];
      b0v.y = b0p[3];
      b1v.x = b1p[0];
      b1v.y = b1p[3];
      acc0 = wmma_f32(a, b0v, acc0);
      acc1 = wmma_f32(a, b1v, acc1);
      a.x = la[oa1];
      a.y = la[ob1];
      b0v.x = b0p[12];
      b0v.y = b0p[15];
      b1v.x = b1p[12];
      b1v.y = b1p[15];
      acc0 = wmma_f32(a, b0v, acc0);
      acc1 = wmma_f32(a, b1v, acc1);
      a.x = la[oa2];
      a.y = la[ob2];
      b0v.x = b0p[24];
      b0v.y = b0p[27];
      b1v.x = b1p[24];
      b1v.y = b1p[27];
      acc0 = wmma_f32(a, b0v, acc0);
      acc1 = wmma_f32(a, b1v, acc1);
      la += 32;
      b0p += 36;
      b1p += 36;
    }
    // Leftover full k-steps (NFULL%3) + partial tail (K%4), generic & clamped.
    for (int k0 = N3 * 12; k0 < K; k0 += 4) {
      const int ka = k0 + lh * 2, kb = ka + 1;
      const int kac = min(ka, K - 1), kbc = min(kb, K - 1);
      const float a0 = lA[(kac / 3) * 8 + (kac % 3)];
      const float a1 = lA[(kbc / 3) * 8 + (kbc % 3)];
      v2f a, b0v, b1v;
      a.x = (ka < K) ? a0 : 0.f;  // zero A beyond K; B stays finite (0*b = 0)
      a.y = (kb < K) ? a1 : 0.f;
      b0v.x = W[(long)nc0 * CIN * 9 + 3 * kac + 1];
      b0v.y = W[(long)nc0 * CIN * 9 + 3 * kbc + 1];
      b1v.x = W[(long)nc1 * CIN * 9 + 3 * kac + 1];
      b1v.y = W[(long)nc1 * CIN * 9 + 3 * kbc + 1];
      acc0 = wmma_f32(a, b0v, acc0);
      acc1 = wmma_f32(a, b1v, acc1);
    }
    // ---- epilogue: bias + relu + store + fused BN partial sums ----
#pragma unroll
    for (int half = 0; half < 2; ++half) {
      const int nn = half ? nn1 : nn0;
      const v8f acc = half ? acc1 : acc0;
      const float bv = bias[half ? nc1 : nc0];  // clamped -> unconditional
      float s = 0.f, s2 = 0.f;
      if (nn < COUT) {
        if (fullT) {  // uniform fast path: no per-row guards
#pragma unroll
          for (int rr = 0; rr < 8; ++rr) {
            const int m2 = mt * 16 + rr + lh * 8;
            const int tl2 = m2 / 6, h2 = m2 % 6;
            const float v = fmaxf(acc[rr] + bv, 0.f);
            out[((long)(t0 + tl2) * COUT + nn) * 6 + h2] = v;
            s += v;
            s2 += v * v;
          }
        } else {
#pragma unroll
          for (int rr = 0; rr < 8; ++rr) {
            const int m2 = mt * 16 + rr + lh * 8;
            const int tl2 = m2 / 6, h2 = m2 % 6;
            const int tg = t0 + tl2;
            const float v = fmaxf(acc[rr] + bv, 0.f);
            if (tg < T) {
              out[((long)tg * COUT + nn) * 6 + h2] = v;
              s += v;
              s2 += v * v;
            }
          }
        }
      }
      if (STATS) {
        s += __shfl_down(s, 16, 32);  // fold the two half-lanes of column nn
        s2 += __shfl_down(s2, 16, 32);
        if (lh == 0 && nn < COUT) {
          atomicAdd(&bsum[nn], s);
          atomicAdd(&bsq[nn], s2);
        }
      }
    }
  }
  if (STATS) {
    __syncthreads();
    for (int i = tid; i < COUT; i += 128) {
      atomicAdd(&gsum[i], bsum[i]);
      atomicAdd(&gsq[i], bsq[i]);
    }
  }
}

// ---------------------------------------------------------------------------
// Kernel 4: BatchNorm (batch stats) + lean Linear(6,6), one thread per (t,n).
// ---------------------------------------------------------------------------
__global__ __launch_bounds__(256) void bn_lean(
    const float* __restrict__ f3, const float* __restrict__ gsum,
    const float* __restrict__ gsq, const float* __restrict__ gamma,
    const float* __restrict__ beta, const float* __restrict__ Wl,
    const float* __restrict__ bl, float* __restrict__ out, long M,
    float invCount) {
  const long idx = (long)blockIdx.x * blockDim.x + threadIdx.x;
  if (idx >= M) return;
  const int nch = (int)(idx % N_NODES);
  const float mean = gsum[nch] * invCount;
  const float var = gsq[nch] * invCount - mean * mean;
  const float sc = gamma[nch] * rsqrtf(var + 1e-5f);
  const float sh = beta[nch] - mean * sc;
  const float* row = f3 + idx * HID;
  float y[HID];
#pragma unroll
  for (int h = 0; h < HID; ++h) y[h] = fmaf(row[h], sc, sh);
#pragma unroll
  for (int j = 0; j < HID; ++j) {
    float o = bl[j];
#pragma unroll
    for (int h = 0; h < HID; ++h) o = fmaf(y[h], Wl[j * HID + h], o);
    out[idx * HID + j] = o;
  }
}

__global__ void zero_f32(float* p, int n) {
  const int i = blockIdx.x * blockDim.x + threadIdx.x;
  if (i < n) p[i] = 0.f;
}

// ---------------------------------------------------------------------------
extern "C" void kernel_launch(void* const* d_in, const int* in_sizes, int n_in,
                              void* d_out, int out_size, void* d_ws,
                              size_t ws_size, hipStream_t stream) {
  (void)n_in;
  (void)out_size;
  (void)ws_size;
  const float* X = (const float*)d_in[1];
  const float* Wih = (const float*)d_in[3];
  const float* Whh = (const float*)d_in[4];
  const float* bih = (const float*)d_in[5];
  const float* bhh = (const float*)d_in[6];
  const float* Wc1 = (const float*)d_in[7];
  const float* bc1 = (const float*)d_in[8];
  const float* Wc2 = (const float*)d_in[9];
  const float* bc2 = (const float*)d_in[10];
  const float* Wc3 = (const float*)d_in[11];
  const float* bc3 = (const float*)d_in[12];
  const float* gamma = (const float*)d_in[13];
  const float* beta = (const float*)d_in[14];
  const float* Wl = (const float*)d_in[15];
  const float* bl = (const float*)d_in[16];
  float* out = (float*)d_out;

  const long T = (long)in_sizes[1] / (N_NODES * IN_T);  // 16384
  const long M = T * N_NODES;

  // workspace layout (floats): xp | al | f1 | f2 | f3 | stats(sum, sumsq)
  float* ws = (float*)d_ws;
  float* xp = ws;                // M*24
  float* al = xp + M * GATES;    // M*6
  float* f1 = al + M * HID;      // T*194*6
  float* f2 = f1 + T * 194 * 6;  // T*194*6
  float* f3 = f2 + T * 194 * 6;  // M*6
  float* gsum = f3 + M * HID;    // 97
  float* gsq = gsum + N_NODES;   // 97

  zero_f32<<<1, 256, 0, stream>>>(gsum, 2 * N_NODES);

  const long mtiles = (M + 15) / 16;
  const unsigned xblocks = (unsigned)((mtiles + 3) / 4);
  xproj_wmma<<<xblocks, 128, 0, stream>>>(X, Wih, bih, xp, M);

  lstm_scan<<<N_NODES, 32, 0, stream>>>(xp, Whh, bhh, al, (int)T);

  const unsigned cblocks = (unsigned)((T + 7) / 8);
  conv3x1_wmma<97, 194, false>
      <<<cblocks, 128, 0, stream>>>(al, Wc1, bc1, f1, nullptr, nullptr, (int)T);
  conv3x1_wmma<194, 194, false>
      <<<cblocks, 128, 0, stream>>>(f1, Wc2, bc2, f2, nullptr, nullptr, (int)T);
  conv3x1_wmma<194, 97, true>
      <<<cblocks, 128, 0, stream>>>(f2, Wc3, bc3, f3, gsum, gsq, (int)T);

  const float invCount = 1.0f / (float)(T * HID);
  const unsigned fblocks = (unsigned)((M + 255) / 256);
  bn_lean<<<fblocks, 256, 0, stream>>>(f3, gsum, gsq, gamma, beta, Wl, bl, out,
                                       M, invCount);
}